// TimeStretchModule_47906065219852
// MI455X (gfx1250) — compile-verified
//
#include <hip/hip_runtime.h>
#include <math.h>

// Phase-vocoder time stretch for MI455X (gfx1250, wave32).
// One wave32 per (batch,freq) row; shuffle-based inclusive scan for the
// sequential phase accumulation; f64 index math to match numpy arange.

#define WAVE 32
#define WAVES_PER_BLOCK 8
#define BLOCK (WAVE * WAVES_PER_BLOCK)

__global__ __launch_bounds__(BLOCK) void pv_timestretch_kernel(
    const float2* __restrict__ x,    // (rows, n_frames) complex as float2
    const float* __restrict__ pa,    // (n_freq) phase advance
    float2* __restrict__ out,        // (rows, n_out) complex as float2
    int rows, int n_freq, int n_frames, int n_out)
{
    const int lane = threadIdx.x & (WAVE - 1);
    const int row  = blockIdx.x * WAVES_PER_BLOCK + (threadIdx.x >> 5);
    if (row >= rows) return;  // row is wave-uniform

    const int   f         = row % n_freq;
    const float phase_adv = pa[f];

    const float2* __restrict__ xr   = x   + (size_t)row * (size_t)n_frames;
    float2*       __restrict__ orow = out + (size_t)row * (size_t)n_out;

    // inc for t==0 is angle(frame 0); carried in prev_w.
    float2 x0    = xr[0];
    float  prev_w = atan2f(x0.y, x0.x);
    float  carry  = 0.0f;

    const float  TWO_PI     = 6.28318530717958647692f;
    const float  INV_TWO_PI = 0.15915494309189533577f;
    const double RATE       = 1.07;

    const int nchunks = (n_out + WAVE - 1) / WAVE;
    for (int c = 0; c < nchunks; ++c) {
        const int t = c * WAVE + lane;
        float w   = 0.0f;   // wrapped phase delta at t (== increment for t+1)
        float mag = 0.0f;
        const bool live = (t < n_out);

        if (live) {
            // Match numpy: time_steps = arange(0, n_frames, 1.07) in float64.
            double ts = (double)t * RATE;
            double fl = floor(ts);
            int    i0 = (int)fl;
            float  alpha = (float)(ts - fl);

            // Reference pads 2 zero frames at the end.
            float2 r0 = (i0     < n_frames) ? xr[i0]     : make_float2(0.0f, 0.0f);
            float2 r1 = (i0 + 1 < n_frames) ? xr[i0 + 1] : make_float2(0.0f, 0.0f);

            float a0 = atan2f(r0.y, r0.x);
            float a1 = atan2f(r1.y, r1.x);
            float n0 = sqrtf(r0.x * r0.x + r0.y * r0.y);
            float n1 = sqrtf(r1.x * r1.x + r1.y * r1.y);
            mag = alpha * n1 + (1.0f - alpha) * n0;

            // wrap(a1 - a0 - pa) + pa ; jnp.round == rint (ties-to-even)
            float d = a1 - a0 - phase_adv;
            d = d - TWO_PI * rintf(d * INV_TWO_PI);
            w = d + phase_adv;
        }

        // inc[t] = (t==0) ? phase_0 : w(t-1)   -> shift by one lane across wave
        float inc = __shfl_up(w, 1, WAVE);
        if (lane == 0) inc = prev_w;

        // Wave32 inclusive scan (5 steps) — lowers to ds_bpermute lane permutes.
        float s = inc;
        #pragma unroll
        for (int off = 1; off < WAVE; off <<= 1) {
            float v = __shfl_up(s, off, WAVE);
            if (lane >= off) s += v;
        }
        const float phase_acc = carry + s;

        // Carry running sum and last wrapped delta to the next chunk.
        carry  += __shfl(s, WAVE - 1, WAVE);
        prev_w  = __shfl(w, WAVE - 1, WAVE);

        if (live) {
            float sn, cs;
            __sincosf(phase_acc, &sn, &cs);
            orow[t] = make_float2(mag * cs, mag * sn);
        }

        // Prefetch input for two chunks ahead (gfx1250 global_prefetch_b8).
        const int tp = t + 2 * WAVE;
        if (tp < n_out) {
            int ip = (int)floor((double)tp * RATE);
            if (ip < n_frames) __builtin_prefetch((const void*)(xr + ip), 0, 0);
        }
    }
}

extern "C" void kernel_launch(void* const* d_in, const int* in_sizes, int n_in,
                              void* d_out, int out_size, void* d_ws, size_t ws_size,
                              hipStream_t stream) {
    (void)n_in; (void)d_ws; (void)ws_size;

    const float2* x  = (const float2*)d_in[0];   // spectrogram (16,513,4096,2) f32
    const float*  pa = (const float*) d_in[1];   // phase_advance (513,1) f32
    float2*       o  = (float2*)d_out;

    const int n_frames = 4096;
    const int n_freq   = in_sizes[1];                       // 513
    const int rows     = in_sizes[0] / (n_frames * 2);      // 16*513 = 8208
    const int n_out    = out_size / (rows * 2);             // 3829

    const int blocks = (rows + WAVES_PER_BLOCK - 1) / WAVES_PER_BLOCK;  // 1026
    pv_timestretch_kernel<<<blocks, BLOCK, 0, stream>>>(x, pa, o, rows, n_freq, n_frames, n_out);
}